// PointNetEx_32409823216140
// MI455X (gfx1250) — compile-verified
//
#include <hip/hip_runtime.h>
#include <hip/hip_bf16.h>

// ---------------------------------------------------------------------------
// CDNA5 (gfx1250) WMMA primitives: full-fp32 matrix core, 16x16 tiles, K=4.
// A (16x4): lane m=lane&15 holds row m; half (lane>=16) selects K pair.
// B (4x16): lane n=lane&15 holds col n; half selects K pair.
// C/D (16x16): vgpr r holds row r (lanes 0-15) / row 8+r (lanes 16-31), col=lane&15.
// ---------------------------------------------------------------------------
typedef float v2f __attribute__((ext_vector_type(2)));
typedef float v8f __attribute__((ext_vector_type(8)));

__device__ __forceinline__ v8f wmma4(v2f a, v2f b, v8f c) {
  return __builtin_amdgcn_wmma_f32_16x16x4_f32(
      /*neg_a=*/false, a, /*neg_b=*/false, b,
      /*c_mod=*/(short)0, c, /*reuse_a=*/false, /*reuse_b=*/false);
}

// Async memory->LDS copy (CDNA5 Tensor/async path, ASYNCcnt-tracked).
// One 16-byte transfer per lane, no VGPR round-trip.
__device__ __forceinline__ void async_ld_b128(float* lds_dst, const float* gsrc) {
  unsigned ldsOff = (unsigned)(uintptr_t)lds_dst;  // low 32 bits = LDS offset
  asm volatile("global_load_async_to_lds_b128 %0, %1, off"
               :: "v"(ldsOff), "v"(gsrc) : "memory");
}
__device__ __forceinline__ void wait_async0() {
  asm volatile("s_wait_asynccnt 0x0" ::: "memory");
}

#define D_MODEL 256
#define D_FF    1024
#define ATT_SCALE 0.25f   // 1/sqrt(16)

// ---------------------------------------------------------------------------
// 256->256 GEMM + bias: Y[T,256] = X[T,256] @ W[256,256] + b
// grid T/16 (1D), block 128 (4 waves). Each wave owns 4 adjacent 16x16 tiles
// (64 cols) so every LDS A-fragment feeds 4 WMMAs. X tile staged in LDS via
// async global->LDS copies.
// ---------------------------------------------------------------------------
__global__ __launch_bounds__(128)
void k_gemm(const float* __restrict__ X, const float* __restrict__ W,
            const float* __restrict__ bias, float* __restrict__ Y) {
  __shared__ __align__(16) float lX[16 * D_MODEL];
  const int tid = threadIdx.x;
  const float* Xt = X + (size_t)blockIdx.x * 16 * D_MODEL;
#pragma unroll
  for (int i = 0; i < 8; i++) {
    const int e = tid + i * 128;                 // 1024 x 16B = 16 KB
    async_ld_b128(&lX[e * 4], Xt + e * 4);
  }
  const int wave = tid >> 5, lane = tid & 31, hi = lane >> 4, l15 = lane & 15;
  const int col0 = wave * 64 + l15;
  __builtin_prefetch(&W[col0], 0, 1);            // weight panel is L2-resident
  wait_async0();
  __syncthreads();

  v8f c0 = {0.f, 0.f, 0.f, 0.f, 0.f, 0.f, 0.f, 0.f};
  v8f c1 = c0, c2 = c0, c3 = c0;
#pragma unroll 4
  for (int kb = 0; kb < D_MODEL; kb += 4) {
    v2f a;
    a.x = lX[l15 * D_MODEL + kb + 2 * hi];
    a.y = lX[l15 * D_MODEL + kb + 2 * hi + 1];
    const size_t r0 = (size_t)(kb + 2 * hi) * D_MODEL + col0;
    const size_t r1 = r0 + D_MODEL;
    v2f b0, b1, b2, b3;
    b0.x = W[r0 +  0]; b0.y = W[r1 +  0];
    b1.x = W[r0 + 16]; b1.y = W[r1 + 16];
    b2.x = W[r0 + 32]; b2.y = W[r1 + 32];
    b3.x = W[r0 + 48]; b3.y = W[r1 + 48];
    c0 = wmma4(a, b0, c0);
    c1 = wmma4(a, b1, c1);
    c2 = wmma4(a, b2, c2);
    c3 = wmma4(a, b3, c3);
  }
  const size_t rowBase = (size_t)blockIdx.x * 16 + 8 * hi;
  const float bv0 = bias[col0], bv1 = bias[col0 + 16];
  const float bv2 = bias[col0 + 32], bv3 = bias[col0 + 48];
#pragma unroll
  for (int r = 0; r < 8; r++) {
    const size_t row = (rowBase + r) * D_MODEL + col0;
    Y[row +  0] = c0[r] + bv0;
    Y[row + 16] = c1[r] + bv1;
    Y[row + 32] = c2[r] + bv2;
    Y[row + 48] = c3[r] + bv3;
  }
}

// ---------------------------------------------------------------------------
// Fused FFN: H[T,256] = relu(X @ W1[256,1024] + b1) @ W2[1024,256] + b2
// grid T/16, block 128, dynamic LDS = 16KB (X tile) + 64KB (hidden tile).
// X tile staged via async global->LDS.
// ---------------------------------------------------------------------------
__global__ __launch_bounds__(128)
void k_ffn(const float* __restrict__ X, const float* __restrict__ W1,
           const float* __restrict__ b1, const float* __restrict__ W2,
           const float* __restrict__ b2, float* __restrict__ H) {
  extern __shared__ float smem[];
  float* lX = smem;                 // 16 x 256
  float* lH = smem + 16 * D_MODEL;  // 16 x 1024
  const int tid = threadIdx.x;
  const float* Xt = X + (size_t)blockIdx.x * 16 * D_MODEL;
#pragma unroll
  for (int i = 0; i < 8; i++) {
    const int e = tid + i * 128;
    async_ld_b128(&lX[e * 4], Xt + e * 4);
  }
  wait_async0();
  __syncthreads();

  const int wave = tid >> 5, lane = tid & 31, hi = lane >> 4, l15 = lane & 15;
  for (int ct = wave; ct < D_FF / 16; ct += 4) {
    const int col = ct * 16 + l15;
    v8f c = {0.f, 0.f, 0.f, 0.f, 0.f, 0.f, 0.f, 0.f};
#pragma unroll 4
    for (int kb = 0; kb < D_MODEL; kb += 4) {
      v2f a, b;
      a.x = lX[l15 * D_MODEL + kb + 2 * hi];
      a.y = lX[l15 * D_MODEL + kb + 2 * hi + 1];
      b.x = W1[(size_t)(kb + 2 * hi) * D_FF + col];
      b.y = W1[(size_t)(kb + 2 * hi + 1) * D_FF + col];
      c = wmma4(a, b, c);
    }
    const float bv = b1[col];
#pragma unroll
    for (int r = 0; r < 8; r++)
      lH[(r + 8 * hi) * D_FF + col] = fmaxf(c[r] + bv, 0.f);
  }
  __syncthreads();
  for (int ct = wave; ct < D_MODEL / 16; ct += 4) {
    const int col = ct * 16 + l15;
    v8f c = {0.f, 0.f, 0.f, 0.f, 0.f, 0.f, 0.f, 0.f};
#pragma unroll 4
    for (int kb = 0; kb < D_FF; kb += 4) {
      v2f a, b;
      a.x = lH[l15 * D_FF + kb + 2 * hi];
      a.y = lH[l15 * D_FF + kb + 2 * hi + 1];
      b.x = W2[(size_t)(kb + 2 * hi) * D_MODEL + col];
      b.y = W2[(size_t)(kb + 2 * hi + 1) * D_MODEL + col];
      c = wmma4(a, b, c);
    }
    const float bv = b2[col];
#pragma unroll
    for (int r = 0; r < 8; r++)
      H[((size_t)blockIdx.x * 16 + r + 8 * hi) * D_MODEL + col] = c[r] + bv;
  }
}

// ---------------------------------------------------------------------------
// Fused masked self-attention over groups of 32 tokens (enc1 / dec).
// grid = 2048 groups, block 128 (4 waves); each wave handles 8 (head,qtile) jobs.
// s = QK^T (WMMA), masked softmax via xor-shuffles in 16-lane halves,
// probability tile goes C-layout -> LDS -> A-layout, o = aV (WMMA).
// ---------------------------------------------------------------------------
__global__ __launch_bounds__(128)
void k_attn32(const float* __restrict__ Q, const float* __restrict__ K,
              const float* __restrict__ V, const int* __restrict__ gidx,
              float* __restrict__ O) {
  __shared__ __align__(16) float scr[4][16 * 32];
  const int g = blockIdx.x;
  const int wave = threadIdx.x >> 5, lane = threadIdx.x & 31;
  const int hi = lane >> 4, l15 = lane & 15;
  float* sc = scr[wave];

  const float m0 = (gidx[g * 32 + l15] >= 0) ? 0.f : -1e9f;
  const float m1 = (gidx[g * 32 + 16 + l15] >= 0) ? 0.f : -1e9f;

  for (int job = wave; job < 32; job += 4) {
    const int h = job >> 1, qt = job & 1;
    const int qr0 = g * 32 + qt * 16;
    v8f s0 = {0.f, 0.f, 0.f, 0.f, 0.f, 0.f, 0.f, 0.f};
    v8f s1 = s0;
#pragma unroll
    for (int kb = 0; kb < 16; kb += 4) {
      v2f a, bk0, bk1;
      a.x = Q[(size_t)(qr0 + l15) * D_MODEL + h * 16 + kb + 2 * hi];
      a.y = Q[(size_t)(qr0 + l15) * D_MODEL + h * 16 + kb + 2 * hi + 1];
      bk0.x = K[(size_t)(g * 32 + l15) * D_MODEL + h * 16 + kb + 2 * hi];
      bk0.y = K[(size_t)(g * 32 + l15) * D_MODEL + h * 16 + kb + 2 * hi + 1];
      bk1.x = K[(size_t)(g * 32 + 16 + l15) * D_MODEL + h * 16 + kb + 2 * hi];
      bk1.y = K[(size_t)(g * 32 + 16 + l15) * D_MODEL + h * 16 + kb + 2 * hi + 1];
      s0 = wmma4(a, bk0, s0);
      s1 = wmma4(a, bk1, s1);
    }
#pragma unroll
    for (int r = 0; r < 8; r++) {
      float v0 = s0[r] * ATT_SCALE + m0;
      float v1 = s1[r] * ATT_SCALE + m1;
      float mx = fmaxf(v0, v1);
      for (int d = 1; d < 16; d <<= 1) mx = fmaxf(mx, __shfl_xor(mx, d));
      float e0 = __expf(v0 - mx), e1 = __expf(v1 - mx);
      float sm = e0 + e1;
      for (int d = 1; d < 16; d <<= 1) sm += __shfl_xor(sm, d);
      const float inv = 1.f / sm;
      sc[(r + 8 * hi) * 32 + l15] = e0 * inv;
      sc[(r + 8 * hi) * 32 + 16 + l15] = e1 * inv;
    }
    v8f o = {0.f, 0.f, 0.f, 0.f, 0.f, 0.f, 0.f, 0.f};
#pragma unroll
    for (int kb = 0; kb < 32; kb += 4) {
      v2f a2, b2;
      a2.x = sc[l15 * 32 + kb + 2 * hi];
      a2.y = sc[l15 * 32 + kb + 2 * hi + 1];
      b2.x = V[(size_t)(g * 32 + kb + 2 * hi) * D_MODEL + h * 16 + l15];
      b2.y = V[(size_t)(g * 32 + kb + 2 * hi + 1) * D_MODEL + h * 16 + l15];
      o = wmma4(a2, b2, o);
    }
#pragma unroll
    for (int r = 0; r < 8; r++)
      O[(size_t)(qr0 + r + 8 * hi) * D_MODEL + h * 16 + l15] = o[r];
  }
}

// ---------------------------------------------------------------------------
// Flash attention for enc2: per (batch, head, 16-row q-tile), stream 64 key
// tiles with online softmax. grid 512, block 128 (1 job per wave).
// ---------------------------------------------------------------------------
__global__ __launch_bounds__(128)
void k_flash(const float* __restrict__ Q, const float* __restrict__ K,
             const float* __restrict__ V, const float* __restrict__ mask,
             float* __restrict__ O) {
  __shared__ __align__(16) float scr[4][16 * 16];
  const int wave = threadIdx.x >> 5, lane = threadIdx.x & 31;
  const int hi = lane >> 4, l15 = lane & 15;
  const int job = blockIdx.x * 4 + wave;          // 2 * 16 * 64 = 2048 jobs
  const int b = job >> 10, rem = job & 1023;
  const int h = rem >> 6, qt = rem & 63;
  const int qr0 = b * 1024 + qt * 16;
  float* sc = scr[wave];

  v8f o = {0.f, 0.f, 0.f, 0.f, 0.f, 0.f, 0.f, 0.f};
  float mrun[8], lrun[8];
#pragma unroll
  for (int r = 0; r < 8; r++) { mrun[r] = -3.4e38f; lrun[r] = 0.f; }

  for (int kt = 0; kt < 64; kt++) {
    const int kr0 = b * 1024 + kt * 16;
    v8f s = {0.f, 0.f, 0.f, 0.f, 0.f, 0.f, 0.f, 0.f};
#pragma unroll
    for (int kb = 0; kb < 16; kb += 4) {
      v2f a, bk;
      a.x = Q[(size_t)(qr0 + l15) * D_MODEL + h * 16 + kb + 2 * hi];
      a.y = Q[(size_t)(qr0 + l15) * D_MODEL + h * 16 + kb + 2 * hi + 1];
      bk.x = K[(size_t)(kr0 + l15) * D_MODEL + h * 16 + kb + 2 * hi];
      bk.y = K[(size_t)(kr0 + l15) * D_MODEL + h * 16 + kb + 2 * hi + 1];
      s = wmma4(a, bk, s);
    }
    const float mk = (1.f - mask[b * 1024 + kt * 16 + l15]) * (-1e9f);
#pragma unroll
    for (int r = 0; r < 8; r++) {
      float sv = s[r] * ATT_SCALE + mk;
      float tm = sv;
      for (int d = 1; d < 16; d <<= 1) tm = fmaxf(tm, __shfl_xor(tm, d));
      const float nm = fmaxf(mrun[r], tm);
      const float f = __expf(mrun[r] - nm);
      const float p = __expf(sv - nm);
      float ps = p;
      for (int d = 1; d < 16; d <<= 1) ps += __shfl_xor(ps, d);
      lrun[r] = lrun[r] * f + ps;
      o[r] *= f;
      mrun[r] = nm;
      sc[(r + 8 * hi) * 16 + l15] = p;
    }
#pragma unroll
    for (int kb = 0; kb < 16; kb += 4) {
      v2f a2, b2;
      a2.x = sc[l15 * 16 + kb + 2 * hi];
      a2.y = sc[l15 * 16 + kb + 2 * hi + 1];
      b2.x = V[(size_t)(kr0 + kb + 2 * hi) * D_MODEL + h * 16 + l15];
      b2.y = V[(size_t)(kr0 + kb + 2 * hi + 1) * D_MODEL + h * 16 + l15];
      o = wmma4(a2, b2, o);
    }
  }
#pragma unroll
  for (int r = 0; r < 8; r++)
    O[(size_t)(qr0 + r + 8 * hi) * D_MODEL + h * 16 + l15] = o[r] / lrun[r];
}

// ---------------------------------------------------------------------------
// Single-query attention (Sq=1), Sk in {1,16,32}. One thread per (token,head).
// knn != null : keyrow = (t>>qshift)*rowsPerB + knn[t*SK+j] (gathered keys)
// knn == null : keyrow = (t>>qshift)*SK + j (contiguous key block)
// maskI != null : key invalid if maskI[keyrow] < 0
// ---------------------------------------------------------------------------
template <int SK>
__global__ __launch_bounds__(256)
void k_attn1q(const float* __restrict__ Q, const float* __restrict__ Kp,
              const float* __restrict__ Vp, const int* __restrict__ knn,
              const int* __restrict__ maskI, float* __restrict__ O,
              int qshift, int rowsPerB) {
  const int job = blockIdx.x * 256 + threadIdx.x;
  const int t = job >> 4, h = job & 15;
  float q[16];
#pragma unroll
  for (int d = 0; d < 16; d++) q[d] = Q[(size_t)t * D_MODEL + h * 16 + d];

  int kr[SK];
  float lg[SK];
  float mx = -3.4e38f;
#pragma unroll
  for (int j = 0; j < SK; j++) {
    int r;
    if (knn) r = (t >> qshift) * rowsPerB + knn[(size_t)t * SK + j];
    else     r = (t >> qshift) * SK + j;
    kr[j] = r;
    float d0 = 0.f;
#pragma unroll
    for (int d = 0; d < 16; d++) d0 += q[d] * Kp[(size_t)r * D_MODEL + h * 16 + d];
    d0 *= ATT_SCALE;
    if (maskI && maskI[r] < 0) d0 += -1e9f;
    lg[j] = d0;
    mx = fmaxf(mx, d0);
  }
  float sum = 0.f;
#pragma unroll
  for (int j = 0; j < SK; j++) { lg[j] = __expf(lg[j] - mx); sum += lg[j]; }
  const float inv = 1.f / sum;
  float acc[16];
#pragma unroll
  for (int d = 0; d < 16; d++) acc[d] = 0.f;
#pragma unroll
  for (int j = 0; j < SK; j++) {
    const float a = lg[j] * inv;
#pragma unroll
    for (int d = 0; d < 16; d++) acc[d] += a * Vp[(size_t)kr[j] * D_MODEL + h * 16 + d];
  }
#pragma unroll
  for (int d = 0; d < 16; d++) O[(size_t)t * D_MODEL + h * 16 + d] = acc[d];
}

// ---------------------------------------------------------------------------
// y = LayerNorm(x + h) * g + b. Block 256 = 8 waves, one token per wave.
// ---------------------------------------------------------------------------
__global__ __launch_bounds__(256)
void k_lnres(const float* __restrict__ X, const float* __restrict__ Hh,
             const float* __restrict__ g, const float* __restrict__ bta,
             float* __restrict__ Y) {
  const int wave = threadIdx.x >> 5, lane = threadIdx.x & 31;
  const size_t t = (size_t)blockIdx.x * 8 + wave;
  float v[8];
  float s = 0.f;
#pragma unroll
  for (int i = 0; i < 8; i++) {
    const int c = lane + 32 * i;
    v[i] = X[t * D_MODEL + c] + Hh[t * D_MODEL + c];
    s += v[i];
  }
  for (int d = 1; d < 32; d <<= 1) s += __shfl_xor(s, d);
  const float mean = s * (1.f / D_MODEL);
  float var = 0.f;
#pragma unroll
  for (int i = 0; i < 8; i++) { const float dd = v[i] - mean; var += dd * dd; }
  for (int d = 1; d < 32; d <<= 1) var += __shfl_xor(var, d);
  const float inv = rsqrtf(var * (1.f / D_MODEL) + 1e-5f);
#pragma unroll
  for (int i = 0; i < 8; i++) {
    const int c = lane + 32 * i;
    Y[t * D_MODEL + c] = (v[i] - mean) * inv * g[c] + bta[c];
  }
}

// --------------------------- small utility kernels -------------------------
__global__ void k_embed(const float* __restrict__ x, const float* __restrict__ w,
                        const float* __restrict__ b, float* __restrict__ y) {
  const int t = blockIdx.x, c = threadIdx.x;   // grid = T, block = 256
  y[(size_t)t * D_MODEL + c] = b[c] + x[t * 3 + 0] * w[c] +
                               x[t * 3 + 1] * w[D_MODEL + c] +
                               x[t * 3 + 2] * w[2 * D_MODEL + c];
}

__global__ void k_gather(const float* __restrict__ XE, const int* __restrict__ gidx,
                         float* __restrict__ XG) {
  const int idx = blockIdx.x * 256 + threadIdx.x;   // 65536 rows * 64 chunks
  const int row = idx >> 6, c4 = (idx & 63) * 4;
  const int raw = gidx[row];
  const int b = (row >> 5) >> 10;
  float4 v = make_float4(0.f, 0.f, 0.f, 0.f);
  if (raw >= 0)
    v = *(const float4*)&XE[((size_t)b * 16384 + raw) * D_MODEL + c4];
  *(float4*)&XG[(size_t)row * D_MODEL + c4] = v;
}

__global__ void k_scatter(const float* __restrict__ XG, const int* __restrict__ gidx,
                          float* __restrict__ ACC, float* __restrict__ CNT) {
  const int idx = blockIdx.x * 256 + threadIdx.x;
  const int row = idx >> 6, c4 = (idx & 63) * 4;
  const int raw = gidx[row];
  const int b = (row >> 5) >> 10;
  const int src = (raw >= 0) ? raw : 0;
  if ((idx & 63) == 0) atomicAdd(&CNT[b * 16384 + src], 1.f);
  if (raw >= 0) {
    const float* s = &XG[(size_t)row * D_MODEL + c4];
    float* d = &ACC[((size_t)b * 16384 + src) * D_MODEL + c4];
    atomicAdd(d + 0, s[0]); atomicAdd(d + 1, s[1]);
    atomicAdd(d + 2, s[2]); atomicAdd(d + 3, s[3]);
  }
}

__global__ void k_middle(const float* __restrict__ ACC, const float* __restrict__ CNT,
                         float* __restrict__ XM) {
  const size_t i = (size_t)blockIdx.x * 256 + threadIdx.x;
  XM[i] = ACC[i] / fmaxf(CNT[i >> 8], 1.f);
}

__global__ __launch_bounds__(256)
void k_outproj(const float* __restrict__ Xo, const float* __restrict__ w,
               const float* __restrict__ b, float* __restrict__ out) {
  const int wave = threadIdx.x >> 5, lane = threadIdx.x & 31;
  const size_t t = (size_t)blockIdx.x * 8 + wave;
  float s = 0.f;
#pragma unroll
  for (int i = 0; i < 8; i++) {
    const int c = lane + 32 * i;
    s += Xo[t * D_MODEL + c] * w[c];
  }
  for (int d = 1; d < 32; d <<= 1) s += __shfl_xor(s, d);
  if (lane == 0) out[t] = s + b[0];
}

// ---------------------------------------------------------------------------
// Host orchestration. Param dict flattened depth-first:
// 0..4 tensors; 5-8 emb_x/emb_xs; 9 enc1(6x16); 105 cross_enc(16);
// 121 enc2(6x16); 217 cross_dec(16); 233 dec(6x16); 329/330 out.
// Per-layer order: qw qb kw kb vw vb ow ob g1 b1 f1w f1b f2w f2b g2 b2
// Workspace requirement: ~482 MiB.
// ---------------------------------------------------------------------------
extern "C" void kernel_launch(void* const* d_in, const int* in_sizes, int n_in,
                              void* d_out, int out_size, void* d_ws, size_t ws_size,
                              hipStream_t stream) {
  (void)in_sizes; (void)n_in; (void)out_size; (void)ws_size;
  const float* x    = (const float*)d_in[0];
  const float* xs0  = (const float*)d_in[1];
  const int*   gidx = (const int*)d_in[2];
  const int*   knn  = (const int*)d_in[3];
  const float* smask= (const float*)d_in[4];
  auto P = [&](int i) { return (const float*)d_in[i]; };

  size_t off = 0;
  auto alloc = [&](size_t elems) -> float* {
    float* p = (float*)((char*)d_ws + off);
    off += ((elems * 4 + 255) / 256) * 256;
    return p;
  };
  float* XE  = alloc(8388608);    // [2,16384,256]
  float* XS  = alloc(524288);     // [2,1024,256]
  float* XG  = alloc(16777216);   // [2048*32,256]
  float* QB  = alloc(16777216);
  float* KB  = alloc(16777216);
  float* VB  = alloc(16777216);
  float* OB  = alloc(16777216);
  float* TB  = alloc(16777216);
  float* ACC = alloc(8388608);
  float* CNT = alloc(32768);
  float* XM  = alloc(8388608);

  auto gemm = [&](const float* Xp, const float* W, const float* bb, float* Y, int T) {
    k_gemm<<<T / 16, 128, 0, stream>>>(Xp, W, bb, Y);
  };
  auto lnres = [&](const float* Xp, const float* Hp, const float* gg,
                   const float* bb, float* Y, int T) {
    k_lnres<<<T / 8, 256, 0, stream>>>(Xp, Hp, gg, bb, Y);
  };
  auto ffn = [&](const float* Xp, int base, float* Hp, int T) {
    k_ffn<<<T / 16, 128, (16 * D_MODEL + 16 * D_FF) * 4, stream>>>(
        Xp, P(base + 10), P(base + 11), P(base + 12), P(base + 13), Hp);
  };
  auto tail = [&](int base, float* Xp, int T) {  // o-proj, ln1, ffn, ln2
    gemm(OB, P(base + 6), P(base + 7), TB, T);
    lnres(Xp, TB, P(base + 8), P(base + 9), Xp, T);
    ffn(Xp, base, TB, T);
    lnres(Xp, TB, P(base + 14), P(base + 15), Xp, T);
  };
  auto self32 = [&](int base) {  // enc1 / dec layer on XG
    const int T = 65536;
    gemm(XG, P(base + 0), P(base + 1), QB, T);
    gemm(XG, P(base + 2), P(base + 3), KB, T);
    gemm(XG, P(base + 4), P(base + 5), VB, T);
    k_attn32<<<2048, 128, 0, stream>>>(QB, KB, VB, gidx, OB);
    tail(base, XG, T);
  };
  auto flashL = [&](int base) {  // enc2 layer on XS
    const int T = 2048;
    gemm(XS, P(base + 0), P(base + 1), QB, T);
    gemm(XS, P(base + 2), P(base + 3), KB, T);
    gemm(XS, P(base + 4), P(base + 5), VB, T);
    k_flash<<<512, 128, 0, stream>>>(QB, KB, VB, smask, OB);
    tail(base, XS, T);
  };

  // 1) embeddings + group gather
  k_embed<<<32768, 256, 0, stream>>>(x, P(5), P(6), XE);
  k_embed<<<2048, 256, 0, stream>>>(xs0, P(7), P(8), XS);
  k_gather<<<16384, 256, 0, stream>>>(XE, gidx, XG);

  // 2) encoder1 (6 layers, groups of 32, mask from group_index)
  for (int i = 0; i < 6; i++) self32(9 + 16 * i);

  // 3) cross-encoder: xs queries xg groups (Sk=32, masked)
  {
    const int base = 105, Tq = 2048;
    gemm(XS, P(base + 0), P(base + 1), QB, Tq);
    gemm(XG, P(base + 2), P(base + 3), KB, 65536);
    gemm(XG, P(base + 4), P(base + 5), VB, 65536);
    k_attn1q<32><<<Tq * 16 / 256, 256, 0, stream>>>(QB, KB, VB, nullptr, gidx, OB, 0, 0);
    tail(base, XS, Tq);
  }

  // 4) encoder2 (6 flash-attention layers over 1024 sampled points)
  for (int i = 0; i < 6; i++) flashL(121 + 16 * i);

  // 5) cross-decoder (mid): xg queries its group's xs token (Sk=1)
  {
    const int base = 217, Tq = 65536;
    gemm(XG, P(base + 0), P(base + 1), QB, Tq);
    gemm(XS, P(base + 2), P(base + 3), KB, 2048);
    gemm(XS, P(base + 4), P(base + 5), VB, 2048);
    k_attn1q<1><<<Tq * 16 / 256, 256, 0, stream>>>(QB, KB, VB, nullptr, nullptr, OB, 5, 0);
    tail(base, XG, Tq);
  }

  // 6) decoder (6 layers)
  for (int i = 0; i < 6; i++) self32(233 + 16 * i);

  // 7) scatter-mean -> x_middle
  hipMemsetAsync(ACC, 0, (size_t)8388608 * 4, stream);
  hipMemsetAsync(CNT, 0, (size_t)32768 * 4, stream);
  k_scatter<<<16384, 256, 0, stream>>>(XG, gidx, ACC, CNT);
  k_middle<<<32768, 256, 0, stream>>>(ACC, CNT, XM);

  // 8) final shared cross-decoder: project x_middle once, gather K/V by knn
  {
    const int base = 217, Tq = 32768;
    gemm(XE, P(base + 0), P(base + 1), QB, Tq);
    gemm(XM, P(base + 2), P(base + 3), KB, Tq);
    gemm(XM, P(base + 4), P(base + 5), VB, Tq);
    k_attn1q<16><<<Tq * 16 / 256, 256, 0, stream>>>(QB, KB, VB, knn, nullptr, OB, 14, 16384);
    tail(base, XE, Tq);
  }

  // 9) output projection [256 -> 1]
  k_outproj<<<4096, 256, 0, stream>>>(XE, P(329), P(330), (float*)d_out);
}